// Capslayer_44547400794256
// MI455X (gfx1250) — compile-verified
//
#include <hip/hip_runtime.h>

// Problem constants from the reference
#define B_     16
#define IN_C   32
#define OUT_C  32
#define KHW    14
#define S_INN  9
#define S_OUTT 9
#define NLOC   (IN_C * KHW * KHW)     // 6272 locations (c,h,w)
#define M_TOT  (OUT_C * S_OUTT)       // 288 rows (o,t)
#define MTILES (M_TOT / 16)           // 18
#define LOCS_PER_WAVE 16
#define NCHUNK (NLOC / LOCS_PER_WAVE) // 392
#define EPSF   1e-8f

typedef float v2f __attribute__((ext_vector_type(2)));
typedef float v8f __attribute__((ext_vector_type(8)));

__global__ void zero_ws_kernel(float* __restrict__ ws, int n) {
    int i = blockIdx.x * blockDim.x + threadIdx.x;
    if (i < n) ws[i] = 0.0f;
}

// One wave per (M-tile, 16-location chunk). grid=(392,18), block=32 (wave32).
__global__ __launch_bounds__(32) void caps_wmma_kernel(
    const float* __restrict__ x,   // (16, 32, 14, 14, 9)
    const float* __restrict__ p,   // (16, 32, 14, 14)
    const float* __restrict__ T,   // (1, 32, 32, 14, 14, 9, 9)
    float* __restrict__ S1,        // (16, 288) accumulators: sum(pred*pf)
    float* __restrict__ S2)        // (16, 288) accumulators: sum(pred^2*pf)
{
    const int lane = threadIdx.x;      // 0..31
    const int half = lane >> 4;        // 0 or 1
    const int l    = lane & 15;
    const int mt   = blockIdx.y;       // 0..17
    // A-matrix row this lane supplies: m = mt*16 + l  ->  (o, t)
    const int mrow = mt * 16 + l;      // < 288 always
    const int o_a  = mrow / S_OUTT;
    const int t_a  = mrow - o_a * S_OUTT;
    const int loc0 = blockIdx.x * LOCS_PER_WAVE;
    const int sb   = 2 * half;

    // Branchless K=8 tail: only lanes 0..15 (half==0) carry the real s=8 value.
    const float msk  = (half == 0) ? 1.0f : 0.0f;
    const float msk1 = 0.1f * msk;

    float s1a[8], s2a[8];
#pragma unroll
    for (int r = 0; r < 8; ++r) { s1a[r] = 0.0f; s2a[r] = 0.0f; }

#pragma unroll 2
    for (int j = 0; j < LOCS_PER_WAVE; ++j) {
        const int loc = loc0 + j;              // uniform across the wave
        const int c   = loc / (KHW * KHW);
        const int hw  = loc - c * (KHW * KHW);

        // Ts[c, o_a, h, w, t_a, s] * 0.1 ; s contiguous
        const float* Arow = T + (((c * OUT_C + o_a) * (KHW * KHW) + hw) * (S_OUTT * S_INN)
                                 + t_a * S_INN);
        // x[b=l, c, h, w, s] ; s contiguous
        const float* Brow = x + ((size_t)l * NLOC + loc) * S_INN;
        const float  pfv  = p[(size_t)l * NLOC + loc];   // column weight, b = l

        // K-step 0: s in {0,1} (half 0) / {2,3} (half 1)
        v2f a0; a0.x = Arow[sb]     * 0.1f; a0.y = Arow[sb + 1] * 0.1f;
        v2f b0; b0.x = Brow[sb];            b0.y = Brow[sb + 1];
        // K-step 1: s in {4,5}/{6,7}
        v2f a1; a1.x = Arow[4 + sb] * 0.1f; a1.y = Arow[5 + sb] * 0.1f;
        v2f b1; b1.x = Brow[4 + sb];        b1.y = Brow[5 + sb];
        // K-step 2: only s=8 is real (half 0, x component); zero-pad via mask,
        // loads are unconditional so no EXEC-divergent branches are generated.
        v2f a2; a2.x = Arow[8] * msk1; a2.y = 0.0f;
        v2f b2; b2.x = Brow[8] * msk;  b2.y = 0.0f;

        v8f d = {};
        d = __builtin_amdgcn_wmma_f32_16x16x4_f32(false, a0, false, b0, (short)0, d, false, false);
        d = __builtin_amdgcn_wmma_f32_16x16x4_f32(false, a1, false, b1, (short)0, d, false, false);
        d = __builtin_amdgcn_wmma_f32_16x16x4_f32(false, a2, false, b2, (short)0, d, false, false);

#pragma unroll
        for (int r = 0; r < 8; ++r) {
            float dv = d[r];
            s1a[r] += dv * pfv;
            s2a[r] += dv * dv * pfv;
        }
    }

    // D layout: (vgpr r, lane) -> row mt*16 + r + 8*half, col b = l
#pragma unroll
    for (int r = 0; r < 8; ++r) {
        const int m   = mt * 16 + r + 8 * half;
        const int idx = l * M_TOT + m;
        atomicAdd(&S1[idx], s1a[r]);
        atomicAdd(&S2[idx], s2a[r]);
    }
}

// Single block: psum, caps, var->vs, per-batch max, p_updated.
__global__ __launch_bounds__(512) void caps_finalize_kernel(
    const float* __restrict__ p,
    const float* __restrict__ S1,
    const float* __restrict__ S2,
    float* __restrict__ out)       // [0,4608): caps  [4608,5120): p_updated
{
    __shared__ float red[512];
    __shared__ float ps_s[16];
    __shared__ float mx_s[16];
    const int tid = threadIdx.x;

    // Phase A: psum[b] = sum_loc p[b, loc]
    {
        const int b = tid & 15;
        const int i = tid >> 4;            // 0..31
        float partial = 0.0f;
        for (int loc = i; loc < NLOC; loc += 32)
            partial += p[(size_t)b * NLOC + loc];
        red[tid] = partial;
    }
    __syncthreads();
    if (tid < 16) {
        float s = 0.0f;
        for (int j = 0; j < 32; ++j) s += red[tid + 16 * j];
        ps_s[tid] = s;
    }
    __syncthreads();

    // Phase B: one thread per (b, o)
    const int b = tid >> 5;   // 0..15
    const int o = tid & 31;   // 0..31
    const float ps    = ps_s[b];
    const float denom = ps + EPSF;
    float vs = 0.0f;
    for (int t = 0; t < S_OUTT; ++t) {
        const int m  = o * S_OUTT + t;
        const float s1 = S1[b * M_TOT + m];
        const float s2 = S2[b * M_TOT + m];
        const float caps = s1 / denom;
        const float var  = (s2 - 2.0f * caps * s1 + caps * caps * ps) / denom;
        out[b * M_TOT + m] = caps;
        vs += var;
    }
    red[tid] = vs;            // red[b*32 + o]
    __syncthreads();
    if (tid < 16) {
        float mx = red[tid * 32];
        for (int j = 1; j < 32; ++j) mx = fmaxf(mx, red[tid * 32 + j]);
        mx_s[tid] = mx;
    }
    __syncthreads();
    out[B_ * M_TOT + b * 32 + o] = 1.0f - vs / (mx_s[b] + EPSF);
}

extern "C" void kernel_launch(void* const* d_in, const int* in_sizes, int n_in,
                              void* d_out, int out_size, void* d_ws, size_t ws_size,
                              hipStream_t stream) {
    (void)in_sizes; (void)n_in; (void)out_size; (void)ws_size;
    const float* x = (const float*)d_in[0];
    const float* p = (const float*)d_in[1];
    // d_in[2] = epoch (unused)
    const float* T = (const float*)d_in[3];
    float* out = (float*)d_out;

    float* S1 = (float*)d_ws;          // 16*288 floats
    float* S2 = S1 + B_ * M_TOT;       // 16*288 floats
    const int nz = 2 * B_ * M_TOT;     // 9216 floats to zero each call

    zero_ws_kernel<<<(nz + 255) / 256, 256, 0, stream>>>(S1, nz);
    caps_wmma_kernel<<<dim3(NCHUNK, MTILES), 32, 0, stream>>>(x, p, T, S1, S2);
    caps_finalize_kernel<<<1, 512, 0, stream>>>(p, S1, S2, out);
}